// NeighborhoodConstraint_27702539059202
// MI455X (gfx1250) — compile-verified
//
#include <hip/hip_runtime.h>
#include <cstdint>

// NeighborhoodConstraint on MI455X (gfx1250, wave32).
// One wave (32 lanes) per row i. Async-gather 32 neighbor rows + x/v rows
// into LDS, compute deltas in place, WMMA (f32 16x16x4) for num = Delta * v,
// VALU/shfl for norms + expm1 weighting, LDS-fed weighted sum for output.

#define Gdim 128
#define Cdim 32
#define RSTR 132   // padded LDS row stride in floats (528 B) to spread banks

typedef __attribute__((ext_vector_type(2))) float v2f;
typedef __attribute__((ext_vector_type(8))) float v8f;

__device__ __forceinline__ float wave_sum32(float x) {
#pragma unroll
  for (int m = 16; m; m >>= 1) x += __shfl_xor(x, m, 32);
  return x;
}

__global__ __launch_bounds__(32)
void NeighborhoodConstraint_27702539059202_kernel(
    const float* __restrict__ x, const float* __restrict__ v,
    const int* __restrict__ k, const float* __restrict__ X,
    float* __restrict__ out)
{
  __shared__ __align__(16) float sdelta[Cdim * RSTR];
  __shared__ __align__(16) float sxrow[Gdim];
  __shared__ __align__(16) float svrow[Gdim];

  const int i    = blockIdx.x;
  const int lane = (int)threadIdx.x;

  // ---------------- async gather: x row, v row, 32 neighbor rows ----------
  // Each instruction moves 32 lanes x 16 B = one full 512 B row into LDS.
  {
    uint32_t ldst = (uint32_t)(uintptr_t)(&sxrow[0]) + (uint32_t)(lane * 16);
    uint64_t gsrc = (uint64_t)(uintptr_t)(x + (size_t)i * Gdim + lane * 4);
    asm volatile("global_load_async_to_lds_b128 %0, %1, off"
                 :: "v"(ldst), "v"(gsrc) : "memory");
    ldst = (uint32_t)(uintptr_t)(&svrow[0]) + (uint32_t)(lane * 16);
    gsrc = (uint64_t)(uintptr_t)(v + (size_t)i * Gdim + lane * 4);
    asm volatile("global_load_async_to_lds_b128 %0, %1, off"
                 :: "v"(ldst), "v"(gsrc) : "memory");
  }
  const uint32_t sdelta_base = (uint32_t)(uintptr_t)(&sdelta[0]);
#pragma unroll 4
  for (int c = 0; c < Cdim; ++c) {
    const int kc = k[(size_t)i * Cdim + c];  // uniform -> scalar load
    uint64_t gsrc = (uint64_t)(uintptr_t)(X + (size_t)kc * Gdim + lane * 4);
    uint32_t ldst = sdelta_base + (uint32_t)(c * RSTR * 4) + (uint32_t)(lane * 16);
    asm volatile("global_load_async_to_lds_b128 %0, %1, off"
                 :: "v"(ldst), "v"(gsrc) : "memory");
  }
  asm volatile("s_wait_asynccnt 0" ::: "memory");

  // ---------------- delta = X[k[c]] - x, in place in LDS ------------------
  const float4 x4 = *(const float4*)&sxrow[lane * 4];
#pragma unroll 4
  for (int c = 0; c < Cdim; ++c) {
    float4* p = (float4*)&sdelta[c * RSTR + lane * 4];
    float4 d = *p;
    d.x -= x4.x; d.y -= x4.y; d.z -= x4.z; d.w -= x4.w;
    *p = d;
  }

  // ---------------- WMMA: num[c] = v . delta[c]  (f32 16x16x4 chain) ------
  // A tile: every row = v chunk (replicated) -> every D row equals num[].
  // B tile: Delta^T chunk; its register image is lane (l&15)'s row at
  // column g0 + 2*(l>>4) -- a natural b64 LDS load.
  const int mrow = lane & 15;
  const int koff = (lane >> 4) * 2;
  v8f acc0 = {0.f, 0.f, 0.f, 0.f, 0.f, 0.f, 0.f, 0.f};
  v8f acc1 = {0.f, 0.f, 0.f, 0.f, 0.f, 0.f, 0.f, 0.f};
#pragma unroll 8
  for (int s = 0; s < 32; ++s) {
    const int g0 = s * 4;
    v2f a  = *(const v2f*)&svrow[g0 + koff];
    v2f b0 = *(const v2f*)&sdelta[mrow * RSTR + g0 + koff];
    v2f b1 = *(const v2f*)&sdelta[(mrow + 16) * RSTR + g0 + koff];
    acc0 = __builtin_amdgcn_wmma_f32_16x16x4_f32(
        false, a, false, b0, (short)0, acc0, false, false);
    acc1 = __builtin_amdgcn_wmma_f32_16x16x4_f32(
        false, a, false, b1, (short)0, acc1, false, false);
  }
  // Lane l owns neighbor c = l. All D rows identical, so element 0 suffices.
  const float num_c = (lane < 16) ? acc0[0] : acc1[0];

  // ---------------- per-neighbor stats (lane c = neighbor c) --------------
  const float4 v4 = *(const float4*)&svrow[lane * 4];
  const float vnorm =
      sqrtf(wave_sum32(v4.x * v4.x + v4.y * v4.y + v4.z * v4.z + v4.w * v4.w));

  float d2 = 0.f;
#pragma unroll 8
  for (int q = 0; q < 32; ++q) {
    const float4 d = *(const float4*)&sdelta[lane * RSTR + q * 4];
    d2 += d.x * d.x + d.y * d.y + d.z * d.z + d.w * d.w;
  }
  const float dnorm = sqrtf(d2);

  const float cs = num_c / fmaxf(vnorm * dnorm, 1e-8f);
  const float tv = expm1f(cs * 10.0f);
  const float S  = wave_sum32(fabsf(tv));   // sum |Tv|
  const float T  = wave_sum32(tv);          // sum Tv
  // Tv/S - (T/(32 S))  ==  (tv - T/32) / S
  const float w  = (tv - T * (1.0f / 32.0f)) / S;

  // ---------------- out[g] = sum_c delta[c][g] * w_c ----------------------
  float4 o = {0.f, 0.f, 0.f, 0.f};
#pragma unroll 4
  for (int c = 0; c < Cdim; ++c) {
    const float wc = __shfl(w, c, 32);
    const float4 d = *(const float4*)&sdelta[c * RSTR + lane * 4];
    o.x += d.x * wc; o.y += d.y * wc; o.z += d.z * wc; o.w += d.w * wc;
  }
  *(float4*)(out + (size_t)i * Gdim + lane * 4) = o;
}

extern "C" void kernel_launch(void* const* d_in, const int* in_sizes, int n_in,
                              void* d_out, int out_size, void* d_ws, size_t ws_size,
                              hipStream_t stream) {
  const float* x = (const float*)d_in[0];
  const float* v = (const float*)d_in[1];
  const int*   k = (const int*)d_in[2];
  const float* X = (const float*)d_in[3];
  float* out = (float*)d_out;
  const int n = in_sizes[0] / Gdim;  // 16384
  NeighborhoodConstraint_27702539059202_kernel<<<n, 32, 0, stream>>>(x, v, k, X, out);
}